// SpatialGCN_29386166239249
// MI455X (gfx1250) — compile-verified
//
#include <hip/hip_runtime.h>
#include <hip/hip_bf16.h>

typedef float v2f __attribute__((ext_vector_type(2)));
typedef float v4f __attribute__((ext_vector_type(4)));
typedef float v8f __attribute__((ext_vector_type(8)));

#define N_  64
#define C_  64
#define F_  64
#define T_  300
#define V_  25
#define VP_ 32
#define TB_ 4   // t-values per block (one per wave); T_ % TB_ == 0
static_assert(T_ % TB_ == 0, "t tiles must divide T exactly");

// ---------------------------------------------------------------------------
// Setup: build padded 32x32 normalized adjacency Ahat[dst][src] in workspace.
// Rows/cols >= 25 stay exactly zero -> they annihilate all padding garbage in
// the main kernel's GEMM2. Duplicate edges accumulate (.at[].add semantics).
// ---------------------------------------------------------------------------
__global__ void build_adj_kernel(const int* __restrict__ ei, int E,
                                 float* __restrict__ Ahat) {
    __shared__ float deg[VP_];
    __shared__ float dinv[VP_];
    const int tid = threadIdx.x;

    if (tid < VP_) deg[tid] = 0.0f;
    __syncthreads();
    for (int e = tid; e < E; e += blockDim.x)
        atomicAdd(&deg[ei[E + e]], 1.0f);          // dst row
    __syncthreads();
    if (tid < V_) deg[tid] += 1.0f;                // self loop
    __syncthreads();
    if (tid < VP_) dinv[tid] = (tid < V_ && deg[tid] > 0.0f) ? rsqrtf(deg[tid]) : 0.0f;
    __syncthreads();

    for (int i = tid; i < VP_ * VP_; i += blockDim.x) Ahat[i] = 0.0f;
    __syncthreads();

    for (int e = tid; e < E; e += blockDim.x) {
        int s = ei[e];                              // src
        int d = ei[E + e];                          // dst
        atomicAdd(&Ahat[d * VP_ + s], dinv[s] * dinv[d]);
    }
    if (tid < V_) atomicAdd(&Ahat[tid * VP_ + tid], dinv[tid] * dinv[tid]);
}

// ---------------------------------------------------------------------------
// Main kernel: out_b = W @ x_b @ Ahat^T + bias, one t per wave.
// LDS: X [64][TB_*25] (raw copy, unpadded) | W [64][64] | Ahat [32][32] | bias
// ---------------------------------------------------------------------------
#define XW   (TB_ * V_)                    // 100 floats per c row
#define XOFF 0
#define WOFF (C_ * XW)                     // 6400
#define AOFF (WOFF + F_ * C_)              // 10496
#define BOFF (AOFF + VP_ * VP_)            // 11520
#define LDSF (BOFF + F_)                   // 11584 floats (~45 KB)

__global__ __launch_bounds__(TB_ * 32) void spatial_gcn_kernel(
        const float* __restrict__ x, const float* __restrict__ w,
        const float* __restrict__ bias, const float* __restrict__ Ahat,
        float* __restrict__ out) {
    __shared__ float lds[LDSF];
    float* Xl = lds + XOFF;
    float* Wl = lds + WOFF;
    float* Al = lds + AOFF;
    float* Bl = lds + BOFF;

    const int tid = threadIdx.x;
    const int n   = blockIdx.y;
    const int t0  = blockIdx.x * TB_;
    const int NT  = TB_ * 32;              // 128 threads

    // ---- Stage x slice: 64 rows x 100 contiguous floats, b128 loads ----
    // global row base (floats): (n*64+c)*7500 + t0*25 ; both 16B aligned.
    for (int j = tid; j < C_ * (XW / 4); j += NT) {      // 1600 float4s
        const int c = j / (XW / 4);
        const int q = j % (XW / 4);
        const v4f val = *reinterpret_cast<const v4f*>(
            x + (long)(n * C_ + c) * (T_ * V_) + t0 * V_ + q * 4);
        *reinterpret_cast<v4f*>(Xl + c * XW + q * 4) = val;
    }
    for (int j = tid; j < (F_ * C_) / 4; j += NT)        // W: 1024 float4s
        reinterpret_cast<v4f*>(Wl)[j] = reinterpret_cast<const v4f*>(w)[j];
    for (int j = tid; j < (VP_ * VP_) / 4; j += NT)      // Ahat: 256 float4s
        reinterpret_cast<v4f*>(Al)[j] = reinterpret_cast<const v4f*>(Ahat)[j];
    if (tid < F_ / 4)
        reinterpret_cast<v4f*>(Bl)[tid] = reinterpret_cast<const v4f*>(bias)[tid];
    __syncthreads();

    const int wv   = tid >> 5;             // wave id 0..3 -> owns t = t0+wv
    const int lane = tid & 31;
    const int half = lane >> 4;
    const int l16  = lane & 15;
    const int t    = t0 + wv;
    const int xb   = wv * V_;              // wave's column base inside X rows

    // ---------------- GEMM1: Z = W (64x64) @ x_b (64x[32]) ----------------
    // N-padding columns 25..31 read spill data (finite); their Z columns are
    // multiplied by Ahat's zero rows in GEMM2, so they never reach the output.
    v8f acc[4][2];
#pragma unroll
    for (int mt = 0; mt < 4; ++mt)
#pragma unroll
        for (int nt = 0; nt < 2; ++nt)
#pragma unroll
            for (int r = 0; r < 8; ++r) acc[mt][nt][r] = 0.0f;

#pragma unroll
    for (int k = 0; k < 16; ++k) {
        const int kk = 4 * k + 2 * half;
        v2f av[4], bv[2];
#pragma unroll
        for (int mt = 0; mt < 4; ++mt) {
            const float* p = &Wl[(mt * 16 + l16) * C_ + kk];
            av[mt][0] = p[0];
            av[mt][1] = p[1];
        }
#pragma unroll
        for (int nt = 0; nt < 2; ++nt) {
            bv[nt][0] = Xl[(kk + 0) * XW + xb + nt * 16 + l16];
            bv[nt][1] = Xl[(kk + 1) * XW + xb + nt * 16 + l16];
        }
#pragma unroll
        for (int mt = 0; mt < 4; ++mt)
#pragma unroll
            for (int nt = 0; nt < 2; ++nt)
                acc[mt][nt] = __builtin_amdgcn_wmma_f32_16x16x4_f32(
                    false, av[mt], false, bv[nt], (short)0, acc[mt][nt],
                    false, false);
    }

    // Z write-back into this wave's own 25 columns only (race-free payload;
    // neighbor over-reads of these columns feed only x0 paths).
#pragma unroll
    for (int nt = 0; nt < 2; ++nt) {
        const int vp = nt * 16 + l16;
        if (vp < V_) {
#pragma unroll
            for (int mt = 0; mt < 4; ++mt)
#pragma unroll
                for (int r = 0; r < 8; ++r)
                    Xl[(mt * 16 + half * 8 + r) * XW + xb + vp] = acc[mt][nt][r];
        }
    }

    // ---------------- GEMM2: out = Z (64x[32]) @ Ahat^T (32x32) + bias ----
    v8f o[4][2];
#pragma unroll
    for (int mt = 0; mt < 4; ++mt)
#pragma unroll
        for (int nt = 0; nt < 2; ++nt)
#pragma unroll
            for (int r = 0; r < 8; ++r)
                o[mt][nt][r] = Bl[mt * 16 + half * 8 + r];

#pragma unroll
    for (int k = 0; k < 8; ++k) {
        const int kk = 4 * k + 2 * half;
        v2f av[4], bv[2];
#pragma unroll
        for (int mt = 0; mt < 4; ++mt) {
            av[mt][0] = Xl[(mt * 16 + l16) * XW + xb + kk + 0];
            av[mt][1] = Xl[(mt * 16 + l16) * XW + xb + kk + 1];
        }
#pragma unroll
        for (int nt = 0; nt < 2; ++nt) {
            // B[k][n] = Ahat^T[v'][v] = Ahat[n][k]; rows k>=25 are zero.
            bv[nt][0] = Al[(nt * 16 + l16) * VP_ + kk + 0];
            bv[nt][1] = Al[(nt * 16 + l16) * VP_ + kk + 1];
        }
#pragma unroll
        for (int mt = 0; mt < 4; ++mt)
#pragma unroll
            for (int nt = 0; nt < 2; ++nt)
                o[mt][nt] = __builtin_amdgcn_wmma_f32_16x16x4_f32(
                    false, av[mt], false, bv[nt], (short)0, o[mt][nt],
                    false, false);
    }

    // ---------------- Store: out[n, f, t, v], mask v<25 ----------
#pragma unroll
    for (int nt = 0; nt < 2; ++nt) {
        const int v = nt * 16 + l16;
        if (v < V_) {
#pragma unroll
            for (int mt = 0; mt < 4; ++mt)
#pragma unroll
                for (int r = 0; r < 8; ++r) {
                    const int f = mt * 16 + half * 8 + r;
                    out[(((long)n * F_ + f) * T_ + t) * V_ + v] = o[mt][nt][r];
                }
        }
    }
}

// ---------------------------------------------------------------------------
extern "C" void kernel_launch(void* const* d_in, const int* in_sizes, int n_in,
                              void* d_out, int out_size, void* d_ws, size_t ws_size,
                              hipStream_t stream) {
    const float* x    = (const float*)d_in[0];
    const float* w    = (const float*)d_in[1];
    const float* bias = (const float*)d_in[2];
    const int*   ei   = (const int*)d_in[3];
    const int E = in_sizes[3] / 2;

    float* Ahat = (float*)d_ws;   // 32*32 floats

    build_adj_kernel<<<1, 64, 0, stream>>>(ei, E, Ahat);

    dim3 grid(T_ / TB_, N_);      // 75 x 64
    spatial_gcn_kernel<<<grid, TB_ * 32, 0, stream>>>(
        x, w, bias, Ahat, (float*)d_out);
}